// Transform_71665824301474
// MI455X (gfx1250) — compile-verified
//
#include <hip/hip_runtime.h>
#include <math.h>

// ---------------- problem constants (match reference) ----------------
#define Hh   320
#define Gg   640
#define Bn   8
#define Ss   800
#define Rr   640
#define HH   (Hh*Hh)       // 102400
#define GH   (Gg*Hh)       // 204800
#define GG   (Gg*Gg)       // 409600
#define Pp   (Ss*Rr)       // 512000
#define TWO_PI 6.28318530717958647692f

typedef __attribute__((ext_vector_type(2))) float v2f;
typedef __attribute__((ext_vector_type(8))) float v8f;

// ---------------- device helpers ----------------
__device__ __forceinline__ float dev_beta() {
    // pi * sqrt((W/OS)^2*(OS-0.5)^2 - 0.8) = pi*sqrt(9*2.25 - 0.8)
    return 3.14159265358979323846f * sqrtf(19.45f);
}

__device__ float i0f(float x) {
    float ax = fabsf(x);
    if (ax < 3.75f) {
        float t = ax * (1.0f / 3.75f);
        t *= t;
        return 1.0f + t * (3.5156229f + t * (3.0899424f + t * (1.2067492f +
               t * (0.2659732f + t * (0.0360768f + t * 0.0045813f)))));
    } else {
        float t = 3.75f / ax;
        return (expf(ax) * rsqrtf(ax)) *
               (0.39894228f + t * (0.01328592f + t * (0.00225319f + t * (-0.00157565f +
                t * (0.00916281f + t * (-0.02057706f + t * (0.02635537f +
                t * (-0.01647633f + t * 0.00392377f))))))));
    }
}

// Kaiser-Bessel kernel value / i0(beta), d in oversampled-grid units, W=6
__device__ __forceinline__ float kb_w(float d, float beta, float inv_i0b) {
    float r = d * (1.0f / 3.0f);
    float arg = fmaxf(1.0f - r * r, 0.0f);
    return i0f(beta * sqrtf(arg)) * inv_i0b;
}

// ---------------- table builders ----------------
// A[m,p] = exp(+2*pi*i*(m-160)(p-160)/320) / sqrt(320)   (centered ortho ifft factor)
__global__ void k_build_A(float* __restrict__ Are, float* __restrict__ Aim) {
    int t = blockIdx.x * 256 + threadIdx.x;
    if (t >= HH) return;
    int m = t / Hh, p = t % Hh;
    int ph = ((m - 160) * (p - 160)) % Hh;          // exact integer phase reduction
    float th = TWO_PI * (float)ph * (1.0f / 320.0f);
    float s, c;
    sincosf(th, &s, &c);
    const float nf = 0.05590169943749474f;          // 1/sqrt(320)
    Are[t] = c * nf;
    Aim[t] = s * nf;
}

// Bm[k,u] = exp(-2*pi*i*k*(u-160)/640)   (zero-pad + ifftshift + forward fft, 640x320)
__global__ void k_build_B(float* __restrict__ Bre, float* __restrict__ Bim) {
    int t = blockIdx.x * 256 + threadIdx.x;
    if (t >= GH) return;
    int k = t / Hh, u = t % Hh;
    int ph = (k * (u - 160)) % Gg;                  // |k*(u-160)| <= 102240, exact in int
    float th = TWO_PI * (float)ph * (1.0f / 640.0f);
    float s, c;
    sincosf(th, &s, &c);
    Bre[t] = c;
    Bim[t] = -s;
}

__global__ void k_build_apod(float* __restrict__ ap) {
    int u = blockIdx.x * 256 + threadIdx.x;
    if (u >= Hh) return;
    float beta = dev_beta();
    float f = ((float)u - 160.0f) * (1.0f / 640.0f);
    float pwf = 3.14159265358979f * 6.0f * f;
    float t = beta * beta - pwf * pwf;
    float z = fmaxf(sqrtf(fabsf(t)), 1e-7f);
    float a = ((t > 0.0f) ? sinhf(z) : sinf(z)) / z;
    float anorm = sinhf(beta) / beta;               // center value (f=0)
    ap[u] = a / anorm;
}

// ---------------- complex GEMM via V_WMMA_F32_16X16X4_F32 ----------------
// C(MxN) = A(MxK) * B, complex, all planes SoA row-major f32.
// BT=false: B is KxN row-major.  BT=true: B is NxK row-major (i.e. C = A * B^T).
// One wave (32 lanes) computes a 16x32 tile (two adjacent 16x16 N-tiles): the
// A fragment is loaded once per K-step and reused for both N-tiles, giving
// 8 WMMAs per 6 v2f loads.
// F32 WMMA forbids A/B NEG modifiers (ISA NEG table: CNeg only), so
// Cr = (Ar*Br) - (Ai*Bi) uses two accumulators subtracted at the end;
// Ci = Ar*Bi + Ai*Br chains one accumulator.
template <bool BT>
__global__ __launch_bounds__(64) void cgemm_wmma(
    const float* __restrict__ Are, const float* __restrict__ Aim,
    const float* __restrict__ Bre, const float* __restrict__ Bim,
    float* __restrict__ Cre, float* __restrict__ Cim,
    int M, int N, int K, int sA, int sB, int sC)
{
    const int lane = threadIdx.x;          // 0..31
    const int hi = lane >> 4;              // 0 or 1
    const int lo = lane & 15;
    const int tp = blockIdx.x * blockDim.y + threadIdx.y;   // N-pair index
    const int tm = blockIdx.y;
    if (tp * 32 >= N) return;              // wave-uniform; EXEC stays all-ones
    const int bz = blockIdx.z;
    Are += (size_t)bz * sA;  Aim += (size_t)bz * sA;
    Bre += (size_t)bz * sB;  Bim += (size_t)bz * sB;
    Cre += (size_t)bz * sC;  Cim += (size_t)bz * sC;

    const int m  = tm * 16 + lo;           // A-fragment row (M = lane%16)
    const int n0 = tp * 32 + lo;           // B-fragment col, N-tile 0
    const int n1 = n0 + 16;                // B-fragment col, N-tile 1

    v8f aP0 = {0.f,0.f,0.f,0.f,0.f,0.f,0.f,0.f};   // Ar*Br  (tile 0)
    v8f aM0 = {0.f,0.f,0.f,0.f,0.f,0.f,0.f,0.f};   // Ai*Bi  (tile 0)
    v8f aI0 = {0.f,0.f,0.f,0.f,0.f,0.f,0.f,0.f};   // Ar*Bi + Ai*Br (tile 0)
    v8f aP1 = {0.f,0.f,0.f,0.f,0.f,0.f,0.f,0.f};
    v8f aM1 = {0.f,0.f,0.f,0.f,0.f,0.f,0.f,0.f};
    v8f aI1 = {0.f,0.f,0.f,0.f,0.f,0.f,0.f,0.f};

#pragma unroll 2
    for (int k0 = 0; k0 < K; k0 += 4) {
        const int ka = k0 + 2 * hi;        // K = j + 2*(lane/16), j = element index
        v2f ar = *(const v2f*)(Are + (size_t)m * K + ka);
        v2f ai = *(const v2f*)(Aim + (size_t)m * K + ka);
        v2f br0, bi0, br1, bi1;
        if (BT) {
            br0 = *(const v2f*)(Bre + (size_t)n0 * K + ka);
            bi0 = *(const v2f*)(Bim + (size_t)n0 * K + ka);
            br1 = *(const v2f*)(Bre + (size_t)n1 * K + ka);
            bi1 = *(const v2f*)(Bim + (size_t)n1 * K + ka);
        } else {
            br0.x = Bre[(size_t)ka * N + n0];
            br0.y = Bre[(size_t)(ka + 1) * N + n0];
            bi0.x = Bim[(size_t)ka * N + n0];
            bi0.y = Bim[(size_t)(ka + 1) * N + n0];
            br1.x = Bre[(size_t)ka * N + n1];
            br1.y = Bre[(size_t)(ka + 1) * N + n1];
            bi1.x = Bim[(size_t)ka * N + n1];
            bi1.y = Bim[(size_t)(ka + 1) * N + n1];
        }
        aP0 = __builtin_amdgcn_wmma_f32_16x16x4_f32(false, ar, false, br0,
                                                    (short)0, aP0, false, false);
        aM0 = __builtin_amdgcn_wmma_f32_16x16x4_f32(false, ai, false, bi0,
                                                    (short)0, aM0, false, false);
        aI0 = __builtin_amdgcn_wmma_f32_16x16x4_f32(false, ar, false, bi0,
                                                    (short)0, aI0, false, false);
        aI0 = __builtin_amdgcn_wmma_f32_16x16x4_f32(false, ai, false, br0,
                                                    (short)0, aI0, false, false);
        aP1 = __builtin_amdgcn_wmma_f32_16x16x4_f32(false, ar, false, br1,
                                                    (short)0, aP1, false, false);
        aM1 = __builtin_amdgcn_wmma_f32_16x16x4_f32(false, ai, false, bi1,
                                                    (short)0, aM1, false, false);
        aI1 = __builtin_amdgcn_wmma_f32_16x16x4_f32(false, ar, false, bi1,
                                                    (short)0, aI1, false, false);
        aI1 = __builtin_amdgcn_wmma_f32_16x16x4_f32(false, ai, false, br1,
                                                    (short)0, aI1, false, false);
    }

#pragma unroll
    for (int r = 0; r < 8; ++r) {          // C/D layout: M = r + 8*(lane/16), N = lane%16
        const int mm = tm * 16 + r + 8 * hi;
        Cre[(size_t)mm * N + n0] = aP0[r] - aM0[r];
        Cim[(size_t)mm * N + n0] = aI0[r];
        Cre[(size_t)mm * N + n1] = aP1[r] - aM1[r];
        Cim[(size_t)mm * N + n1] = aI1[r];
    }
}

// ---------------- deterministic two-stage mean/std reduction ----------------
// stage 1: 400 blocks x 256 threads per batch -> partials[b][400][4]
__global__ void k_reduce1(const float* __restrict__ re, const float* __restrict__ im,
                          float* __restrict__ part) {
    __shared__ float s0[256], s1[256], s2[256], s3[256];
    const int b = blockIdx.y;
    const int tid = threadIdx.x;
    const int idx = blockIdx.x * 256 + tid;          // 400*256 == 102400 exactly
    float r = re[(size_t)b * HH + idx];
    float i = im[(size_t)b * HH + idx];
    s0[tid] = r; s1[tid] = i; s2[tid] = r * r; s3[tid] = i * i;
    __syncthreads();
    for (int off = 128; off > 0; off >>= 1) {
        if (tid < off) {
            s0[tid] += s0[tid + off]; s1[tid] += s1[tid + off];
            s2[tid] += s2[tid + off]; s3[tid] += s3[tid + off];
        }
        __syncthreads();
    }
    if (tid == 0) {
        float* o = part + ((size_t)b * 400 + blockIdx.x) * 4;
        o[0] = s0[0]; o[1] = s1[0]; o[2] = s2[0]; o[3] = s3[0];
    }
}

// stage 2: one block per batch sums the 400 partials
__global__ void k_reduce2(const float* __restrict__ part, float* __restrict__ stats) {
    __shared__ float s[4][256];
    const int b = blockIdx.x;
    const int tid = threadIdx.x;
    float a0 = 0.f, a1 = 0.f, a2 = 0.f, a3 = 0.f;
    for (int j = tid; j < 400; j += 256) {
        const float* p = part + ((size_t)b * 400 + j) * 4;
        a0 += p[0]; a1 += p[1]; a2 += p[2]; a3 += p[3];
    }
    s[0][tid] = a0; s[1][tid] = a1; s[2][tid] = a2; s[3][tid] = a3;
    __syncthreads();
    for (int off = 128; off > 0; off >>= 1) {
        if (tid < off) {
            s[0][tid] += s[0][tid + off]; s[1][tid] += s[1][tid + off];
            s[2][tid] += s[2][tid + off]; s[3][tid] += s[3][tid + off];
        }
        __syncthreads();
    }
    if (tid == 0) {
        stats[b * 4 + 0] = s[0][0]; stats[b * 4 + 1] = s[1][0];
        stats[b * 4 + 2] = s[2][0]; stats[b * 4 + 3] = s[3][0];
    }
}

// normalize (per batch, real/imag separately, ddof=1) then divide by apod outer-product
__global__ void k_norm_apod(const float* __restrict__ img_re, const float* __restrict__ img_im,
                            const float* __restrict__ stats, const float* __restrict__ ap,
                            float* __restrict__ x_re, float* __restrict__ x_im) {
    int t = blockIdx.x * 256 + threadIdx.x;
    if (t >= Bn * HH) return;
    int b = t / HH;
    int uv = t % HH;
    int u = uv / Hh, v = uv % Hh;
    const float Nf = (float)HH;
    float mr = stats[b * 4 + 0] / Nf;
    float mi = stats[b * 4 + 1] / Nf;
    float vr = (stats[b * 4 + 2] - Nf * mr * mr) / (Nf - 1.0f);
    float vi = (stats[b * 4 + 3] - Nf * mi * mi) / (Nf - 1.0f);
    float isr = rsqrtf(fmaxf(vr, 1e-30f));
    float isi = rsqrtf(fmaxf(vi, 1e-30f));
    float inva = 1.0f / (ap[u] * ap[v]);
    x_re[t] = (img_re[t] - mr) * isr * inva;
    x_im[t] = (img_im[t] - mi) * isi * inva;
}

// ---------------- 6x6 Kaiser-Bessel gather ----------------
__global__ void k_interp(const float* __restrict__ kx, const float* __restrict__ ky,
                         const float* __restrict__ Xre, const float* __restrict__ Xim,
                         float* __restrict__ out) {
    int p = blockIdx.x * 256 + threadIdx.x;
    if (p >= Pp) return;
    const float beta = dev_beta();
    const float inv_i0b = 1.0f / i0f(beta);
    float gx = kx[p] * 2.0f;                 // traj * G/H
    float gy = ky[p] * 2.0f;
    int ix0 = (int)floorf(gx) - 2;           // W/2 - 1 = 2
    int iy0 = (int)floorf(gy) - 2;
    float wx[6], wy[6];
    int ix[6], iy[6];
#pragma unroll
    for (int j = 0; j < 6; ++j) {
        int ixu = ix0 + j;
        int iyu = iy0 + j;
        wx[j] = kb_w(gx - (float)ixu, beta, inv_i0b);
        wy[j] = kb_w(gy - (float)iyu, beta, inv_i0b);
        ix[j] = ((ixu % Gg) + Gg) % Gg;
        iy[j] = ((iyu % Gg) + Gg) % Gg;
    }
    for (int b = 0; b < Bn; ++b) {
        const float* xr = Xre + (size_t)b * GG;
        const float* xi = Xim + (size_t)b * GG;
        float ar = 0.0f, ai = 0.0f;
#pragma unroll
        for (int jx = 0; jx < 6; ++jx) {
            const int ro = ix[jx] * Gg;
            const float wxx = wx[jx];
#pragma unroll
            for (int jy = 0; jy < 6; ++jy) {
                const float w = wxx * wy[jy];
                const int o = ro + iy[jy];
                ar += w * xr[o];
                ai += w * xi[o];
            }
        }
        size_t oidx = 2 * ((size_t)b * Pp + p);
        out[oidx + 0] = ar;
        out[oidx + 1] = ai;
    }
}

// ---------------- launcher ----------------
extern "C" void kernel_launch(void* const* d_in, const int* in_sizes, int n_in,
                              void* d_out, int out_size, void* d_ws, size_t ws_size,
                              hipStream_t stream) {
    (void)in_sizes; (void)n_in; (void)out_size; (void)ws_size;
    const float* ksr = (const float*)d_in[0];   // (8,1,320,320)
    const float* ksi = (const float*)d_in[1];   // (8,1,320,320)
    const float* tkx = (const float*)d_in[2];   // (800,640)
    const float* tky = (const float*)d_in[3];   // (800,640)
    float* out = (float*)d_out;                 // (8,1,800,640) complex -> interleaved f32
    float* ws = (float*)d_ws;

    // workspace layout (floats)
    size_t o = 0;
    float* Are  = ws + o; o += HH;
    float* Aim  = ws + o; o += HH;
    float* Bre  = ws + o; o += GH;
    float* Bim  = ws + o; o += GH;
    float* ap   = ws + o; o += Hh;
    float* stats= ws + o; o += Bn * 4;
    float* part = ws + o; o += (size_t)Bn * 400 * 4;
    float* T1re = ws + o; o += (size_t)Bn * HH;
    float* T1im = ws + o; o += (size_t)Bn * HH;
    float* Imre = ws + o; o += (size_t)Bn * HH;
    float* Imim = ws + o; o += (size_t)Bn * HH;
    float* T2re = ws + o; o += (size_t)Bn * GH;
    float* T2im = ws + o; o += (size_t)Bn * GH;
    float* Xore = ws + o; o += (size_t)Bn * GG;
    float* Xoim = ws + o; o += (size_t)Bn * GG;
    // ximg reuses T1 (T1 is dead once Im* is computed)
    float* Xgre = T1re;
    float* Xgim = T1im;

    // tables
    k_build_A   <<<(HH + 255) / 256, 256, 0, stream>>>(Are, Aim);
    k_build_B   <<<(GH + 255) / 256, 256, 0, stream>>>(Bre, Bim);
    k_build_apod<<<(Hh + 255) / 256, 256, 0, stream>>>(ap);

    dim3 blk(32, 2);  // 2 waves per block, one 16x32 tile-pair each

    // T1 = A(320x320) * Ksp(320x320)         [per batch]
    cgemm_wmma<false><<<dim3(5, 20, Bn), blk, 0, stream>>>(
        Are, Aim, ksr, ksi, T1re, T1im, Hh, Hh, Hh, 0, HH, HH);
    // img = T1 * A^T                          (A symmetric, centered ortho ifft2 done)
    cgemm_wmma<true><<<dim3(5, 20, Bn), blk, 0, stream>>>(
        T1re, T1im, Are, Aim, Imre, Imim, Hh, Hh, Hh, HH, 0, HH);

    // mean/std (ddof=1) per batch, then normalize + apodization divide -> ximg
    k_reduce1<<<dim3(400, Bn), 256, 0, stream>>>(Imre, Imim, part);
    k_reduce2<<<Bn, 256, 0, stream>>>(part, stats);
    k_norm_apod<<<(Bn * HH + 255) / 256, 256, 0, stream>>>(
        Imre, Imim, stats, ap, Xgre, Xgim);

    // T2 = Bm(640x320) * ximg(320x320)
    cgemm_wmma<false><<<dim3(5, 40, Bn), blk, 0, stream>>>(
        Bre, Bim, Xgre, Xgim, T2re, T2im, Gg, Hh, Hh, 0, HH, GH);
    // Xos = T2(640x320) * Bm^T  -> oversampled 640x640 k-space grid
    cgemm_wmma<true><<<dim3(10, 40, Bn), blk, 0, stream>>>(
        T2re, T2im, Bre, Bim, Xore, Xoim, Gg, Gg, Hh, GH, 0, GG);

    // 6x6 KB interpolation at 512000 trajectory points, all 8 batches per thread
    k_interp<<<(Pp + 255) / 256, 256, 0, stream>>>(tkx, tky, Xore, Xoim, out);
}